// EnhancedDiffusionLayer_46368466928018
// MI455X (gfx1250) — compile-verified
//
#include <hip/hip_runtime.h>

#define B_  16
#define C_  8
#define S_  128
#define HW  (S_ * S_)
#define EPSF 1e-6f
#define NSTEPS 10
#define TPW 4            // coupling tiles per wave

typedef __attribute__((ext_vector_type(2))) float v2f;
typedef __attribute__((ext_vector_type(8))) float v8f;

// v_rcp_f32 + one Newton-Raphson step: ~0.5 ulp, short dependency chain
// (vs. the IEEE div expansion that would sit on the Thomas recurrence).
__device__ __forceinline__ float fast_rcp(float x) {
    float r = __builtin_amdgcn_rcpf(x);
    return r * (2.0f - x * r);
}

__device__ __forceinline__ float sigmoidf(float x) { return fast_rcp(1.0f + expf(-x)); }

// ---------------------------------------------------------------------------
// Fused channel coupling + content factors.
//   uc[b,c,p] = sum_d K[c,d] * u[b,d,p]        (V_WMMA_F32_16X16X4_F32 x2)
//   cfu [b,p] = 1 + 0.1*(mean_c sigmoid(u[b,c,p])  - 0.5)
//   cfuc[b,p] = 1 + 0.1*(mean_c sigmoid(uc[b,c,p]) - 0.5)
// One wave = one 16-position tile per loop iteration. Lane l / l+16 jointly
// hold all 8 u-channels of position n (B-matrix layout), so cfu needs one
// cross-lane shuffle; lanes 0-15 hold all 8 uc-channels in D VGPRs 0-7.
// ---------------------------------------------------------------------------
__global__ __launch_bounds__(256) void couple_cf_kernel(const float* __restrict__ u,
                                                        const float* __restrict__ cpl,
                                                        float* __restrict__ uc,
                                                        float* __restrict__ cfu,
                                                        float* __restrict__ cfuc) {
    int wave = (blockIdx.x * blockDim.x + threadIdx.x) >> 5;
    int lane = threadIdx.x & 31;
    int m  = lane & 15;                    // A row (output channel)
    int k0 = (lane >> 4) * 2;              // lanes 0-15 -> K{0,1}; 16-31 -> K{2,3}

    v2f a0, a1;                            // A-matrix, rows 8..15 zero-padded
    a0.x = (m < C_) ? cpl[m * C_ + k0 + 0] : 0.0f;
    a0.y = (m < C_) ? cpl[m * C_ + k0 + 1] : 0.0f;
    a1.x = (m < C_) ? cpl[m * C_ + 4 + k0 + 0] : 0.0f;
    a1.y = (m < C_) ? cpl[m * C_ + 4 + k0 + 1] : 0.0f;

    int n = lane & 15;                     // B column (spatial position in tile)

    for (int it = 0; it < TPW; ++it) {
        int tile = wave * TPW + it;
        int b    = tile >> 10;             // HW/16 = 1024 tiles per batch image
        int posb = (tile & 1023) << 4;

        const float* ub = u + (size_t)b * C_ * HW + posb + n;
        v2f b0, b1;
        b0.x = ub[(size_t)(k0 + 0) * HW];
        b0.y = ub[(size_t)(k0 + 1) * HW];
        b1.x = ub[(size_t)(4 + k0 + 0) * HW];
        b1.y = ub[(size_t)(4 + k0 + 1) * HW];

        v8f c = {};
        c = __builtin_amdgcn_wmma_f32_16x16x4_f32(false, a0, false, b0, (short)0, c, false, false);
        c = __builtin_amdgcn_wmma_f32_16x16x4_f32(false, a1, false, b1, (short)0, c, false, false);

        // partial sigmoid sum over the 4 u-channels this lane holds
        float p = sigmoidf(b0.x) + sigmoidf(b0.y) + sigmoidf(b1.x) + sigmoidf(b1.y);
        float pHi = __shfl_down(p, 16, 32);     // lanes 0-15 get lanes 16-31's half

        if (lane < 16) {
            float* ob = uc + (size_t)b * C_ * HW + posb + n;
            float s2 = 0.0f;
            #pragma unroll
            for (int j = 0; j < C_; ++j) {
                ob[(size_t)j * HW] = c[j];
                s2 += sigmoidf(c[j]);
            }
            size_t ci = (size_t)b * HW + posb + n;
            cfu[ci]  = 1.0f + 0.1f * ((p + pHi) * (1.0f / C_) - 0.5f);
            cfuc[ci] = 1.0f + 0.1f * (s2 * (1.0f / C_) - 0.5f);
        }
    }
}

// ---------------------------------------------------------------------------
// Thomas solve along W (x-sweep), in place. One wave per block, one thread per
// row. LDS slot-reuse: stage coeff/d coalesced -> forward pass overwrites with
// c*/d* -> backward overwrites d* with x -> coalesced writeback.
// Odd stride 129 => conflict-free in all three phases (64 banks, wave32).
// ---------------------------------------------------------------------------
__global__ __launch_bounds__(32) void sweep_x_kernel(float* __restrict__ u,
                                                     const float* __restrict__ base,
                                                     const float* __restrict__ tcoef,
                                                     const float* __restrict__ tquad,
                                                     const float* __restrict__ cf,
                                                     const float* __restrict__ bwraw,
                                                     float t, float scale) {
    __shared__ float sc[32][129];          // coeff -> c*
    __shared__ float sd[32][129];          // d     -> d* -> x
    int tid  = threadIdx.x;
    int row0 = blockIdx.x * 32;
    float t2 = t * t;

    // --- coalesced staging: compute coeff tile + copy d tile ---
    for (int r = 0; r < 32; ++r) {
        int row = row0 + r;                // row = (b*C + c)*S + h
        int h  = row & (S_ - 1);
        int bc = row >> 7;
        int c  = bc & (C_ - 1);
        int b  = bc >> 3;
        const float* pb  = base  + ((size_t)c * S_ + h) * S_;
        const float* ptc = tcoef + ((size_t)c * S_ + h) * S_;
        const float* ptq = tquad + ((size_t)c * S_ + h) * S_;
        const float* pcf = cf    + ((size_t)b * S_ + h) * S_;
        const float* ur  = u     + (size_t)row * S_;
        #pragma unroll
        for (int ch = 0; ch < 4; ++ch) {
            int w = ch * 32 + tid;
            sc[r][w] = fminf(fmaxf((pb[w] + ptc[w] * t + ptq[w] * t2) * pcf[w], EPSF), 5.0f) * scale;
            sd[r][w] = ur[w];
        }
    }
    __syncthreads();

    // --- serial Thomas on this thread's row ---
    float w_lo = sigmoidf(bwraw[3]);       // left
    float w_hi = sigmoidf(bwraw[1]);       // right
    float cp = 0.0f, dp = 0.0f;
    for (int w = 0; w < S_; ++w) {
        float coeff = sc[tid][w];
        float bi = (w == 0)      ? 1.0f + coeff * w_lo
                 : (w == S_ - 1) ? 1.0f + coeff * w_hi
                                 : 1.0f + 2.0f * coeff;
        float ai  = -coeff;
        float inv = fast_rcp(bi - ai * cp + EPSF);
        cp = ai * inv;
        dp = (sd[tid][w] - ai * dp) * inv;
        sc[tid][w] = cp;
        sd[tid][w] = dp;
    }
    float x = 0.0f;
    for (int w = S_ - 1; w >= 0; --w) {
        x = sd[tid][w] - sc[tid][w] * x;
        sd[tid][w] = x;
    }
    __syncthreads();

    // --- coalesced writeback ---
    for (int r = 0; r < 32; ++r) {
        float* ur = u + (size_t)(row0 + r) * S_;
        #pragma unroll
        for (int ch = 0; ch < 4; ++ch) ur[ch * 32 + tid] = sd[r][ch * 32 + tid];
    }
}

// ---------------------------------------------------------------------------
// Thomas solve along H (y-sweep), in place. Lanes = 32 consecutive w -> every
// global access is already a coalesced 128B wave transaction.
// ---------------------------------------------------------------------------
__global__ __launch_bounds__(32) void sweep_y_kernel(float* __restrict__ u,
                                                     const float* __restrict__ base,
                                                     const float* __restrict__ tcoef,
                                                     const float* __restrict__ tquad,
                                                     const float* __restrict__ cf,
                                                     const float* __restrict__ bwraw,
                                                     float t, float scale) {
    __shared__ float cs[32][129];
    __shared__ float ds[32][129];
    int tid = threadIdx.x;
    int wchunk = blockIdx.x & 3;           // S/32 = 4 chunks of w
    int bc = blockIdx.x >> 2;
    int c  = bc & (C_ - 1);
    int b  = bc >> 3;
    int w  = wchunk * 32 + tid;

    float w_lo = sigmoidf(bwraw[0]);       // top
    float w_hi = sigmoidf(bwraw[2]);       // bottom

    float*       ucol = u     + (size_t)bc * HW + w;
    const float* pb   = base  + (size_t)c * HW + w;
    const float* ptc  = tcoef + (size_t)c * HW + w;
    const float* ptq  = tquad + (size_t)c * HW + w;
    const float* pcf  = cf    + (size_t)b * HW + w;
    float t2 = t * t;

    float cp = 0.0f, dp = 0.0f;
    for (int h = 0; h < S_; ++h) {
        size_t o = (size_t)h * S_;
        float coeff = fminf(fmaxf((pb[o] + ptc[o] * t + ptq[o] * t2) * pcf[o], EPSF), 5.0f) * scale;
        float bi = (h == 0)      ? 1.0f + coeff * w_lo
                 : (h == S_ - 1) ? 1.0f + coeff * w_hi
                                 : 1.0f + 2.0f * coeff;
        float ai  = -coeff;
        float inv = fast_rcp(bi - ai * cp + EPSF);
        cp = ai * inv;
        dp = (ucol[o] - ai * dp) * inv;
        cs[tid][h] = cp;
        ds[tid][h] = dp;
    }
    float x = 0.0f;
    for (int h = S_ - 1; h >= 0; --h) {
        x = ds[tid][h] - cs[tid][h] * x;
        ucol[(size_t)h * S_] = x;
    }
}

// ---------------------------------------------------------------------------
extern "C" void kernel_launch(void* const* d_in, const int* in_sizes, int n_in,
                              void* d_out, int out_size, void* d_ws, size_t ws_size,
                              hipStream_t stream) {
    const float* u0  = (const float*)d_in[0];
    const float* ab  = (const float*)d_in[1];
    const float* bb  = (const float*)d_in[2];
    const float* atc = (const float*)d_in[3];
    const float* btc = (const float*)d_in[4];
    const float* atq = (const float*)d_in[5];
    const float* btq = (const float*)d_in[6];
    const float* cpl = (const float*)d_in[7];
    const float* bw  = (const float*)d_in[8];
    float* out = (float*)d_out;

    const size_t ubytes  = (size_t)B_ * C_ * HW * sizeof(float);   // 8 MB
    const size_t cfbytes = (size_t)B_ * HW * sizeof(float);        // 1 MB
    char* ws = (char*)d_ws;
    float* A    = (float*)(ws);
    float* Bbuf = (float*)(ws + ubytes);
    float* cfu  = (float*)(ws + 2 * ubytes);
    float* cfuc = (float*)(ws + 2 * ubytes + cfbytes);

    hipMemcpyAsync(A, u0, ubytes, hipMemcpyDeviceToDevice, stream);

    const float sx = 0.0005f;   // DT/2 / DX^2
    const float sy = 0.001f;    // DT   / DY^2
    double t = 0.0;
    for (int step = 0; step < NSTEPS; ++step) {
        // 16384 tiles / (8 waves * TPW) = 512 blocks
        couple_cf_kernel<<<512, 256, 0, stream>>>(A, cpl, Bbuf, cfu, cfuc);
        // x half-step: alpha(t), cf from u
        sweep_x_kernel<<<512, 32, 0, stream>>>(Bbuf, ab, atc, atq, cfu, bw, (float)t, sx);
        // y full-step: beta(t+dt/2), cf from uc
        sweep_y_kernel<<<512, 32, 0, stream>>>(Bbuf, bb, btc, btq, cfuc, bw, (float)(t + 0.0005), sy);
        // x half-step: alpha(t+dt), cf from uc
        sweep_x_kernel<<<512, 32, 0, stream>>>(Bbuf, ab, atc, atq, cfuc, bw, (float)(t + 0.001), sx);
        float* tmp = A; A = Bbuf; Bbuf = tmp;
        t += 0.001;
    }
    hipMemcpyAsync(out, A, ubytes, hipMemcpyDeviceToDevice, stream);
}